// HungarianMatcher_77824807403797
// MI455X (gfx1250) — compile-verified
//
#include <hip/hip_runtime.h>

// Problem shape (fixed by the reference): B=8, T=16, N=1024 preds, M=128 targets,
// C=32 classes, SD=10 state dims.
#define B_   8
#define T_   16
#define N_   1024
#define M_   128
#define C_   32
#define SD_  10
#define BT_  (B_ * T_)

#define CLASS_W 1.0f
#define POS_W   5.0f
#define VEL_W   2.0f

// padded row stride (dwords) for the LDS probs matrix: odd -> conflict-free
// for 16 consecutive-n lane reads (stride 33 mod 64 banks never repeats in 16).
#define PSTRIDE 33

typedef __attribute__((ext_vector_type(2))) float v2f;
typedef __attribute__((ext_vector_type(8))) float v8f;

// ---------------------------------------------------------------------------
// Kernel 1: build cost[bt][m][n] (row-major M x N per slice == the transposed
// matrix the Hungarian solve wants). 64 MiB total -> L2-resident (192 MB L2).
//
// Matrix-pipe usage per 16x16 tile (wave32, EXEC all-ones throughout):
//  * class cost  : 8x V_WMMA_F32_16X16X4_F32 chained over K=C=32 with a one-hot
//                  A matrix (onehot[m][c] = (tc[m]==c)). Exactly one nonzero
//                  product per output -> bit-exact gather of probs[n][tc[m]].
//  * velocity L2 : 1x V_WMMA_F32_16X16X4_F32 for the cross term pv.tv
//                  (K = 3 velocity dims zero-padded to 4), then
//                  ||pv-tv||^2 = ||pv||^2 + ||tv||^2 - 2 pv.tv.
//
// f32 WMMA operand striping (ISA 7.12.2, wave32):
//   A (16x4, MxK): lanes 0-15 -> M=lane,    v0=K0, v1=K1
//                  lanes16-31 -> M=lane-16, v0=K2, v1=K3
//   B (4x16, KxN): mirrored (lanes 0-15 -> N, v0=K0 row, v1=K1; hi half K2/K3)
//   D (16x16)    : VGPR r, lane l -> M = r + (l>=16 ? 8 : 0), N = l & 15
// ---------------------------------------------------------------------------
__global__ __launch_bounds__(256)
void cost_kernel(const float* __restrict__ logits,       // [BT][N][C]
                 const float* __restrict__ pred_states,  // [BT][N][SD]
                 const int*   __restrict__ tcls_g,       // [BT][M]
                 const float* __restrict__ tgt_states,   // [BT][M][SD]
                 float* __restrict__ cost_ws)            // [BT][M][N]
{
    const int bt  = blockIdx.x;
    const int tid = threadIdx.x;

    __shared__ float probs_s[N_ * PSTRIDE];   // 132 KB: softmax probs, padded
    __shared__ float ppos[N_][3];
    __shared__ float pvel[N_][3];
    __shared__ float pvel2s[N_];
    __shared__ float tpos_s[M_][3];
    __shared__ float tvel_s[M_][3];
    __shared__ float tvel2_s[M_];
    __shared__ int   tclss[M_];

    // ---- Phase 1: softmax probs + pred pos/vel into LDS ----
    for (int n = tid; n < N_; n += 256) {
        const float* lg = logits + ((size_t)bt * N_ + n) * C_;
        if (n + 256 < N_)
            __builtin_prefetch(lg + 256 * C_, 0, 0);   // global_prefetch_b8
        float mx = lg[0];
        #pragma unroll
        for (int c = 1; c < C_; ++c) mx = fmaxf(mx, lg[c]);
        float s = 0.0f;
        float* pr = &probs_s[n * PSTRIDE];
        #pragma unroll
        for (int c = 0; c < C_; ++c) {
            float e = __expf(lg[c] - mx);
            pr[c] = e;
            s += e;
        }
        const float inv = 1.0f / s;
        #pragma unroll
        for (int c = 0; c < C_; ++c) pr[c] *= inv;

        const float* ps = pred_states + ((size_t)bt * N_ + n) * SD_;
        float p0 = ps[0], p1 = ps[1], p2 = ps[2];
        float v0 = ps[3], v1 = ps[4], v2 = ps[5];
        ppos[n][0] = p0; ppos[n][1] = p1; ppos[n][2] = p2;
        pvel[n][0] = v0; pvel[n][1] = v1; pvel[n][2] = v2;
        pvel2s[n] = v0 * v0 + v1 * v1 + v2 * v2;
    }
    for (int m = tid; m < M_; m += 256) {
        const float* ts = tgt_states + ((size_t)bt * M_ + m) * SD_;
        float p0 = ts[0], p1 = ts[1], p2 = ts[2];
        float v0 = ts[3], v1 = ts[4], v2 = ts[5];
        tpos_s[m][0] = p0; tpos_s[m][1] = p1; tpos_s[m][2] = p2;
        tvel_s[m][0] = v0; tvel_s[m][1] = v1; tvel_s[m][2] = v2;
        tvel2_s[m] = v0 * v0 + v1 * v1 + v2 * v2;
        tclss[m]   = tcls_g[bt * M_ + m];
    }
    __syncthreads();

    // ---- Phase 2: wave w owns m-tile w (8 waves x 16 rows = 128 targets),
    //      loops over the 64 n-tiles. All A fragments are loop-invariant. ----
    const int wave = tid >> 5;
    const int lane = tid & 31;
    const int half = lane >> 4;   // lo half: K pair (0,1) ; hi half: K pair (2,3)
    const int l15  = lane & 15;

    // velocity A fragment (targets x K, K=3 padded to 4)
    const int ma = wave * 16 + l15;
    v2f a_vel;
    a_vel.x = half ? tvel_s[ma][2] : tvel_s[ma][0];
    a_vel.y = half ? 0.0f          : tvel_s[ma][1];

    // one-hot class A fragments, chunk q covers K = 4q .. 4q+3
    const int tc_a = tclss[ma];
    v2f a_cls[8];
    #pragma unroll
    for (int q = 0; q < 8; ++q) {
        const int k0 = 4 * q + (half ? 2 : 0);
        a_cls[q].x = (tc_a == k0)     ? 1.0f : 0.0f;
        a_cls[q].y = (tc_a == k0 + 1) ? 1.0f : 0.0f;
    }

    // per-r target constants (row m_d = wave*16 + r + half*8), loop-invariant
    float tr0[8], tr1[8], tr2[8], tv2r[8];
    #pragma unroll
    for (int r = 0; r < 8; ++r) {
        const int m = wave * 16 + r + half * 8;
        tr0[r] = tpos_s[m][0];
        tr1[r] = tpos_s[m][1];
        tr2[r] = tpos_s[m][2];
        tv2r[r] = tvel2_s[m];
    }

    for (int nt = 0; nt < N_ / 16; ++nt) {
        const int n = nt * 16 + l15;

        // velocity cross-term tile
        v2f b_vel;
        b_vel.x = half ? pvel[n][2] : pvel[n][0];
        b_vel.y = half ? 0.0f       : pvel[n][1];
        v8f acc_v = {};
        acc_v = __builtin_amdgcn_wmma_f32_16x16x4_f32(
                    false, a_vel, false, b_vel, (short)0, acc_v, false, false);

        // class tile: D = onehot(16x32) x probs^T(32x16), K=32 as 8 chunks of 4
        const float* pr = &probs_s[n * PSTRIDE];
        v8f acc_c = {};
        #pragma unroll
        for (int q = 0; q < 8; ++q) {
            const int k0 = 4 * q + (half ? 2 : 0);
            v2f b;
            b.x = pr[k0];
            b.y = pr[k0 + 1];
            acc_c = __builtin_amdgcn_wmma_f32_16x16x4_f32(
                        false, a_cls[q], false, b, (short)0, acc_c, false, false);
        }

        const float pv2 = pvel2s[n];
        const float px0 = ppos[n][0], px1 = ppos[n][1], px2 = ppos[n][2];
        #pragma unroll
        for (int r = 0; r < 8; ++r) {
            const int m = wave * 16 + r + half * 8;
            float d2   = tv2r[r] + pv2 - 2.0f * acc_v[r];
            float velc = sqrtf(fmaxf(d2, 0.0f));
            float posc = fabsf(px0 - tr0[r]) +
                         fabsf(px1 - tr1[r]) +
                         fabsf(px2 - tr2[r]);
            cost_ws[((size_t)bt * M_ + m) * N_ + n] =
                -CLASS_W * acc_c[r] + POS_W * posc + VEL_W * velc;
        }
    }
}

// ---------------------------------------------------------------------------
// Kernel 2: Jonker-Volgenant Hungarian, one workgroup per (b,t) slice.
// Serial outer structure (replicated across threads via shared scalars),
// O(n) inner passes parallelized across 256 threads. Cost rows stream
// from L2 (whole cost tensor is L2-resident).
// ---------------------------------------------------------------------------
#define HBS   256
#define INF_F 1e30f

__global__ __launch_bounds__(HBS)
void hungarian_kernel(const float* __restrict__ cost_ws,        // [BT][M][N]
                      const unsigned char* __restrict__ vmask,  // [BT][M] bool
                      int* __restrict__ outp)                   // [BT][M]
{
    const int bt  = blockIdx.x;
    const int tid = threadIdx.x;
    const int n   = N_;

    __shared__ float vsh [N_ + 1];
    __shared__ float minv[N_ + 1];
    __shared__ int   psh [N_ + 1];
    __shared__ int   wsh [N_ + 1];
    __shared__ unsigned char used[N_ + 1];
    __shared__ float ush [M_ + 1];
    __shared__ float rval[HBS];
    __shared__ int   ridx[HBS];
    __shared__ int   sh_j1;
    __shared__ float sh_delta;
    __shared__ int   sh_k;

    if (tid == 0) {
        int c = 0;
        for (int m = 0; m < M_; ++m) c += vmask[bt * M_ + m] ? 1 : 0;
        sh_k = c;
    }
    for (int j = tid; j <= n; j += HBS) { vsh[j] = 0.0f; psh[j] = 0; wsh[j] = 0; }
    for (int m = tid; m <= M_; m += HBS) ush[m] = 0.0f;
    for (int m = tid; m <  M_; m += HBS) outp[bt * M_ + m] = -1;
    __syncthreads();

    const int k = sh_k;
    const float* cost_bt = cost_ws + (size_t)bt * M_ * N_;

    for (int i = 1; i <= k; ++i) {
        for (int j = tid; j <= n; j += HBS) { minv[j] = INF_F; used[j] = 0; }
        if (tid == 0) psh[0] = i;
        __syncthreads();

        int j0 = 0;
        while (true) {
            if (tid == 0) used[j0] = 1;
            __syncthreads();
            const int   i0  = psh[j0];
            const float ui0 = ush[i0];
            const float* crow = cost_bt + (size_t)(i0 - 1) * N_;

            // cur = cost[i0-1][:] - u[i0] - v ; relax minv/way over free cols
            for (int j = tid + 1; j <= n; j += HBS) {
                if (!used[j]) {
                    float cur = crow[j - 1] - ui0 - vsh[j];
                    if (cur < minv[j]) { minv[j] = cur; wsh[j] = j0; }
                }
            }
            __syncthreads();

            // argmin over free columns (first-occurrence tie-break, like np.argmin)
            float bv = INF_F; int bj = n + 1;
            for (int j = tid + 1; j <= n; j += HBS) {
                if (!used[j]) { float mv = minv[j]; if (mv < bv) { bv = mv; bj = j; } }
            }
            rval[tid] = bv; ridx[tid] = bj;
            __syncthreads();
            for (int s = HBS / 2; s > 0; s >>= 1) {
                if (tid < s) {
                    float ov = rval[tid + s]; int oj = ridx[tid + s];
                    if (ov < rval[tid] || (ov == rval[tid] && oj < ridx[tid])) {
                        rval[tid] = ov; ridx[tid] = oj;
                    }
                }
                __syncthreads();
            }
            if (tid == 0) { sh_j1 = ridx[0]; sh_delta = rval[0]; }
            __syncthreads();
            const int   j1    = sh_j1;
            const float delta = sh_delta;

            // delta updates; psh[j] distinct across used j -> race-free scatter
            for (int j = tid; j <= n; j += HBS) {
                if (used[j]) { ush[psh[j]] += delta; vsh[j] -= delta; }
                else if (j >= 1) minv[j] -= delta;
            }
            __syncthreads();

            j0 = j1;
            if (psh[j0] == 0) break;
        }

        // augment along the alternating path
        if (tid == 0) {
            int jj = j0;
            while (jj != 0) { int jp = wsh[jj]; psh[jj] = psh[jp]; jj = jp; }
        }
        __syncthreads();
    }

    // emit: column j assigned to row p[j] -> assign[row-1] = j-1
    for (int j = tid + 1; j <= n; j += HBS) {
        int r = psh[j];
        if (r > 0) outp[bt * M_ + (r - 1)] = j - 1;
    }
}

// ---------------------------------------------------------------------------
extern "C" void kernel_launch(void* const* d_in, const int* in_sizes, int n_in,
                              void* d_out, int out_size, void* d_ws, size_t ws_size,
                              hipStream_t stream)
{
    (void)in_sizes; (void)n_in; (void)out_size; (void)ws_size;

    const float*         logits  = (const float*)d_in[0];         // [B,T,N,C] f32
    const float*         pstates = (const float*)d_in[1];         // [B,T,N,10] f32
    const int*           tcls    = (const int*)d_in[2];           // [B,T,M] i32
    const float*         tstates = (const float*)d_in[3];         // [B,T,M,10] f32
    const unsigned char* vmask   = (const unsigned char*)d_in[4]; // [B,T,M] bool

    int*   outp    = (int*)d_out;    // int32 assign[B,T,M]
    float* cost_ws = (float*)d_ws;   // 64 MiB cost tensor (L2-resident)

    cost_kernel<<<BT_, 256, 0, stream>>>(logits, pstates, tcls, tstates, cost_ws);
    hungarian_kernel<<<BT_, 256, 0, stream>>>(cost_ws, vmask, outp);
}